// CGNN_70566312673786
// MI455X (gfx1250) — compile-verified
//
#include <hip/hip_runtime.h>

typedef __attribute__((ext_vector_type(2))) float v2f;
typedef __attribute__((ext_vector_type(8))) float v8f;

#define D_FEAT 128

__device__ __forceinline__ int iminc(int a, int b) { return a < b ? a : b; }

// ---------- degree accumulation: deg[row[e]] += cv[e] ----------
__global__ void k_degree(const int* __restrict__ row, const float* __restrict__ cv,
                         float* __restrict__ deg, int E) {
    int e = blockIdx.x * blockDim.x + threadIdx.x;
    if (e < E) atomicAdd(&deg[row[e]], cv[e]);
}

// ---------- deg^-1/2 ----------
__global__ void k_dinv(const float* __restrict__ deg, float* __restrict__ dinv, int N) {
    int i = blockIdx.x * blockDim.x + threadIdx.x;
    if (i < N) {
        float d = deg[i];
        dinv[i] = (d > 0.0f) ? rsqrtf(d) : 0.0f;
    }
}

// ---------- normalized edge weights: w = dinv[row]*cv*dinv[col] ----------
__global__ void k_edgew(const int* __restrict__ row, const int* __restrict__ col,
                        const float* __restrict__ cv, const float* __restrict__ dinv,
                        float* __restrict__ w, int E) {
    int e = blockIdx.x * blockDim.x + threadIdx.x;
    if (e < E) w[e] = dinv[row[e]] * cv[e] * dinv[col[e]];
}

// ---------- SpMM scatter: agg[row[e],:] += w[e] * h[col[e],:]  (D=128) ----------
// One wave32 per edge; each lane handles a float4 (128 floats / 32 lanes).
// Feature matrix (51 MB) fits in MI455X's 192 MB L2, so gathers + f32 atomics stay on-chip.
__global__ __launch_bounds__(256) void k_spmm(const int* __restrict__ row,
                                              const int* __restrict__ col,
                                              const float* __restrict__ w,
                                              const float* __restrict__ h,
                                              float* __restrict__ agg, int E) {
    long long gid = (long long)blockIdx.x * blockDim.x + threadIdx.x;
    int e = (int)(gid >> 5);
    if (e >= E) return;
    int lane = (int)(gid & 31);
    int r = row[e];
    int c = col[e];
    float wv = w[e];
    float4 v = *((const float4*)(h + (size_t)c * D_FEAT) + lane);
    float* dst = agg + (size_t)r * D_FEAT + lane * 4;
    atomicAdd(dst + 0, wv * v.x);
    atomicAdd(dst + 1, wv * v.y);
    atomicAdd(dst + 2, wv * v.z);
    atomicAdd(dst + 3, wv * v.w);
}

// ---------- GEMM1: out = relu(A[N,128] @ W[128,128]^T + bias), WMMA f32 16x16x4 ----------
// blockDim = (32, 8): wave y computes 16x16 output tile at (blockIdx.x*16, y*16).
// A-layout (16x4 f32): lane L holds M=L%16, K = k0 + 2*(L/16) + {0,1}
// B-layout (4x16 f32): lane L holds N=L%16, K = k0 + 2*(L/16) + {0,1}   (B[k][n] = W[n][k])
// C/D-layout: VGPR r -> M = r + 8*(L/16), N = L%16
__global__ __launch_bounds__(256) void k_gemm_relu(const float* __restrict__ A,
                                                   const float* __restrict__ W,
                                                   const float* __restrict__ bias,
                                                   float* __restrict__ out, int N) {
    const int m0   = blockIdx.x * 16;
    const int n0   = threadIdx.y * 16;
    const int lane = threadIdx.x;
    const int l15  = lane & 15;
    const int kh   = (lane >> 4) << 1;  // 0 or 2

    const int mA = iminc(m0 + l15, N - 1);           // clamp (keeps EXEC all-ones)
    const float* __restrict__ Ar = A + (size_t)mA * 128 + kh;
    const float* __restrict__ Br = W + (size_t)(n0 + l15) * 128 + kh;

    v8f c = {};
#pragma unroll
    for (int k0 = 0; k0 < 128; k0 += 4) {
        v2f a, b;
        a.x = Ar[k0];
        a.y = Ar[k0 + 1];
        b.x = Br[k0];
        b.y = Br[k0 + 1];
        c = __builtin_amdgcn_wmma_f32_16x16x4_f32(false, a, false, b, (short)0, c,
                                                  false, false);
    }

    const float bv = bias[n0 + l15];
    const int mbase = m0 + ((lane >> 4) << 3);
#pragma unroll
    for (int r = 0; r < 8; ++r) {
        int m = mbase + r;
        if (m < N) {
            float v = c[r] + bv;
            out[(size_t)m * 128 + (n0 + l15)] = fmaxf(v, 0.0f);
        }
    }
}

// ---------- GEMM2: out = A[N,128] @ W[C,128]^T + bias, C=40 (partial last tile) ----------
// blockDim = (32, ceil(C/16)). Out-of-range B columns are zero-scaled so EXEC stays
// all-ones through every WMMA; only the stores are guarded.
__global__ __launch_bounds__(96) void k_gemm_out(const float* __restrict__ A,
                                                 const float* __restrict__ W,
                                                 const float* __restrict__ bias,
                                                 float* __restrict__ out, int N, int C) {
    const int m0   = blockIdx.x * 16;
    const int n0   = threadIdx.y * 16;
    const int lane = threadIdx.x;
    const int l15  = lane & 15;
    const int kh   = (lane >> 4) << 1;

    const int n      = n0 + l15;
    const int mA     = iminc(m0 + l15, N - 1);
    const float bsc  = (n < C) ? 1.0f : 0.0f;
    const float* __restrict__ Ar = A + (size_t)mA * 128 + kh;
    const float* __restrict__ Br = W + (size_t)iminc(n, C - 1) * 128 + kh;

    v8f c = {};
#pragma unroll
    for (int k0 = 0; k0 < 128; k0 += 4) {
        v2f a, b;
        a.x = Ar[k0];
        a.y = Ar[k0 + 1];
        b.x = bsc * Br[k0];
        b.y = bsc * Br[k0 + 1];
        c = __builtin_amdgcn_wmma_f32_16x16x4_f32(false, a, false, b, (short)0, c,
                                                  false, false);
    }

    if (n < C) {
        const float bv = bias[n];
        const int mbase = m0 + ((lane >> 4) << 3);
#pragma unroll
        for (int r = 0; r < 8; ++r) {
            int m = mbase + r;
            if (m < N) out[(size_t)m * C + n] = c[r] + bv;
        }
    }
}

extern "C" void kernel_launch(void* const* d_in, const int* in_sizes, int n_in,
                              void* d_out, int out_size, void* d_ws, size_t ws_size,
                              hipStream_t stream) {
    const float* x  = (const float*)d_in[0];
    const int*   ei = (const int*)d_in[1];  // [2,E] flat: rows then cols
    const float* cv = (const float*)d_in[2];
    const float* W1 = (const float*)d_in[3];
    const float* b1 = (const float*)d_in[4];
    const float* W2 = (const float*)d_in[5];
    const float* b2 = (const float*)d_in[6];

    const int D = D_FEAT;                 // 128
    const int N = in_sizes[0] / D;        // 100000
    const int E = in_sizes[2];            // 1600000
    const int H = in_sizes[4];            // 128
    const int C = in_sizes[6];            // 40

    const int* row = ei;
    const int* col = ei + E;

    // workspace layout (floats): deg[N] | dinv[N] | w[E] | agg[N*D] | h1[N*D]
    float* ws   = (float*)d_ws;
    float* deg  = ws;
    float* dinv = deg + N;
    float* w    = dinv + N;
    float* agg  = w + E;                  // reused for layer-2 aggregation
    float* h1   = agg + (size_t)N * D;

    const int TB = 256;
    const int gE = (E + TB - 1) / TB;
    const int gN = (N + TB - 1) / TB;
    const int gSp = (int)(((long long)E * 32 + TB - 1) / TB);
    const int gTile = (N + 15) / 16;

    // 1) degrees
    hipMemsetAsync(deg, 0, (size_t)N * sizeof(float), stream);
    k_degree<<<gE, TB, 0, stream>>>(row, cv, deg, E);
    // 2) deg^-1/2
    k_dinv<<<gN, TB, 0, stream>>>(deg, dinv, N);
    // 3) normalized edge weights
    k_edgew<<<gE, TB, 0, stream>>>(row, col, cv, dinv, w, E);

    // 4) layer 1: SpMM -> WMMA GEMM + bias + ReLU
    hipMemsetAsync(agg, 0, (size_t)N * D * sizeof(float), stream);
    k_spmm<<<gSp, TB, 0, stream>>>(row, col, w, x, agg, E);
    k_gemm_relu<<<dim3(gTile), dim3(32, H / 16), 0, stream>>>(agg, W1, b1, h1, N);

    // 5) layer 2: SpMM -> WMMA GEMM + bias
    hipMemsetAsync(agg, 0, (size_t)N * D * sizeof(float), stream);
    k_spmm<<<gSp, TB, 0, stream>>>(row, col, w, h1, agg, E);
    k_gemm_out<<<dim3(gTile), dim3(32, (C + 15) / 16), 0, stream>>>(agg, W2, b2,
                                                                    (float*)d_out, N, C);
}